// SOP_1726576855136
// MI455X (gfx1250) — compile-verified
//
#include <hip/hip_runtime.h>
#include <math.h>

// ---------------------------------------------------------------------------
// SOP max-pooled second-order + sign-preserving matrix sqrt + L2 normalize
//   x : [B=16, 1, N=2048, D=64] fp32   ->  out : [B, 4096] fp32
// Per batch:
//   M[i,j] = max_n x[n,i]*x[n,j]                        (VALU, LDS-tiled)
//   S      = matrix-sign(M/||M||_F)                     (NS sign, WMMA f32)
//   A      = S @ M  (= |M|, SPD)                        (WMMA f32)
//   Y      ~ sqrt(A/||A||_F)                            (coupled NS, WMMA f32)
//   sq     = sqrt(||A||_F) * (S @ Y)   == sign-sqrt(M)  (WMMA f32)
//   out    = sq / max(||sq||_F, 1e-12)
// ---------------------------------------------------------------------------

#define LP 66                 // LDS row pitch in floats (66 -> conflict-free column walks)
#define LDS_MAT (64 * LP)
#define SIGN_ITERS 30
#define SQRT_ITERS 22

typedef __attribute__((ext_vector_type(2))) float v2f;
typedef __attribute__((ext_vector_type(8))) float v8f;

// C-tile (tm,tn) of 64x64 product A(64x64) @ B(64x64), both in LDS with pitch LP.
// Fragment layouts per CDNA5 ISA 7.12.2:
//   A 16x4 f32 : lanes 0-15 rows M=0..15 with {vgpr0,vgpr1}={K,K+1}; lanes 16-31 -> K+2,K+3
//   B 4x16 f32 : vgpr i holds row K=i (lanes 0-15) / K=i+2 (lanes 16-31), N = lane%16
__device__ __forceinline__ v8f wmma_mm_tile(const float* __restrict__ A,
                                            const float* __restrict__ B,
                                            int tm, int tn, int lane) {
  const int lmod = lane & 15;
  const int lh   = lane >> 4;                       // 0: lanes 0-15, 1: lanes 16-31
  const float* arow = A + (tm * 16 + lmod) * LP;    // per-lane row of A
  const float* bcol = B + tn * 16 + lmod;           // per-lane column of B
  v8f c = {0.f, 0.f, 0.f, 0.f, 0.f, 0.f, 0.f, 0.f};
#pragma unroll
  for (int k0 = 0; k0 < 64; k0 += 4) {
    v2f a, b;
    a.x = arow[k0 + 2 * lh + 0];
    a.y = arow[k0 + 2 * lh + 1];
    b.x = bcol[(k0 + 2 * lh + 0) * LP];
    b.y = bcol[(k0 + 2 * lh + 1) * LP];
    c = __builtin_amdgcn_wmma_f32_16x16x4_f32(false, a, false, b, (short)0, c,
                                              false, false);
  }
  return c;
}

// C/D layout: vgpr r, lanes 0-15 -> (M = tm*16 + r,     N = tn*16 + lane)
//                    lanes 16-31 -> (M = tm*16 + r + 8, N = tn*16 + lane-16)
__device__ __forceinline__ void store_tile(float* __restrict__ D, int tm, int tn,
                                           int lane, v8f c) {
  const int lmod = lane & 15;
  const int lh   = lane >> 4;
  float* p = D + (tm * 16 + lh * 8) * LP + tn * 16 + lmod;
#pragma unroll
  for (int r = 0; r < 8; ++r) p[r * LP] = c[r];
}

// S <- 1.5*S - 0.5*Q at this tile's positions (in place; tiles are disjoint)
__device__ __forceinline__ void sign_update_tile(float* __restrict__ S, int tm,
                                                 int tn, int lane, v8f q) {
  const int lmod = lane & 15;
  const int lh   = lane >> 4;
  float* p = S + (tm * 16 + lh * 8) * LP + tn * 16 + lmod;
#pragma unroll
  for (int r = 0; r < 8; ++r) p[r * LP] = 1.5f * p[r * LP] - 0.5f * q[r];
}

// T <- 1.5*I - 0.5*W at this tile
__device__ __forceinline__ void store_T_tile(float* __restrict__ T, int tm, int tn,
                                             int lane, v8f w) {
  const int lmod  = lane & 15;
  const int lh    = lane >> 4;
  const int rbase = tm * 16 + lh * 8;
  const int col   = tn * 16 + lmod;
  float* p = T + rbase * LP + col;
#pragma unroll
  for (int r = 0; r < 8; ++r)
    p[r * LP] = ((rbase + r) == col ? 1.5f : 0.0f) - 0.5f * w[r];
}

__global__ __launch_bounds__(256) void sop_signsqrt_kernel(
    const float* __restrict__ x, float* __restrict__ out, int nrows) {
  __shared__ float sS[LDS_MAT];   // sign iterate S
  __shared__ float sT[LDS_MAT];   // temp: x-tile | P=S*S | M | T
  __shared__ float sY[LDS_MAT];   // A, then Y
  __shared__ float sZ[LDS_MAT];   // Z
  __shared__ float sRed[256];
  __shared__ float sScal[2];

  const int tid  = threadIdx.x;
  const int lane = tid & 31;
  const int wave = tid >> 5;
  const int bat  = blockIdx.x;
  const float* xb = x + (size_t)bat * (size_t)nrows * 64;

  const int t0 = wave * 2, t1 = t0 + 1;   // this wave's two C-tiles (of 16)

  // ---------------- Phase A: M = max_n outer(x_n, x_n), 4x4 per thread -------
  const int ti = (tid >> 4) * 4;   // row block
  const int tj = (tid & 15) * 4;   // col block
  float acc[4][4];
#pragma unroll
  for (int a = 0; a < 4; ++a)
#pragma unroll
    for (int q = 0; q < 4; ++q) acc[a][q] = -INFINITY;

  for (int n0 = 0; n0 < nrows; n0 += 64) {
#pragma unroll
    for (int e = 0; e < 16; ++e) {
      int flat = e * 256 + tid;
      int r = flat >> 6, cc = flat & 63;
      sT[r * LP + cc] = xb[(size_t)(n0 + r) * 64 + cc];
    }
    __syncthreads();
#pragma unroll 4
    for (int n = 0; n < 64; ++n) {
      float xi[4], xj[4];
#pragma unroll
      for (int a = 0; a < 4; ++a) xi[a] = sT[n * LP + ti + a];
#pragma unroll
      for (int q = 0; q < 4; ++q) xj[q] = sT[n * LP + tj + q];
#pragma unroll
      for (int a = 0; a < 4; ++a)
#pragma unroll
        for (int q = 0; q < 4; ++q)
          acc[a][q] = fmaxf(acc[a][q], xi[a] * xj[q]);
    }
    __syncthreads();
  }

  // ---------------- normM and S0 = M / ||M||_F --------------------------------
  {
    float ss = 0.f;
#pragma unroll
    for (int a = 0; a < 4; ++a)
#pragma unroll
      for (int q = 0; q < 4; ++q) ss += acc[a][q] * acc[a][q];
    sRed[tid] = ss;
    __syncthreads();
    if (tid == 0) {
      float s = 0.f;
      for (int i = 0; i < 256; ++i) s += sRed[i];
      float nm = sqrtf(s);
      sScal[0] = (nm > 0.f) ? (1.0f / nm) : 0.0f;
    }
    __syncthreads();
    float rnm = sScal[0];
#pragma unroll
    for (int a = 0; a < 4; ++a)
#pragma unroll
      for (int q = 0; q < 4; ++q)
        sS[(ti + a) * LP + tj + q] = acc[a][q] * rnm;
    __syncthreads();
  }

  // ---------------- Newton-Schulz sign iteration: S <- (3S - S^3)/2 ----------
  for (int it = 0; it < SIGN_ITERS; ++it) {
    v8f p0 = wmma_mm_tile(sS, sS, t0 >> 2, t0 & 3, lane);   // P = S@S
    v8f p1 = wmma_mm_tile(sS, sS, t1 >> 2, t1 & 3, lane);
    store_tile(sT, t0 >> 2, t0 & 3, lane, p0);
    store_tile(sT, t1 >> 2, t1 & 3, lane, p1);
    __syncthreads();
    v8f q0 = wmma_mm_tile(sS, sT, t0 >> 2, t0 & 3, lane);   // Q = S@P
    v8f q1 = wmma_mm_tile(sS, sT, t1 >> 2, t1 & 3, lane);
    __syncthreads();
    sign_update_tile(sS, t0 >> 2, t0 & 3, lane, q0);        // S = 1.5S - 0.5Q
    sign_update_tile(sS, t1 >> 2, t1 & 3, lane, q1);
    __syncthreads();
  }

  // ---------------- A = S @ M  (= |M|, SPD); normA; Y0 = A/normA, Z0 = I -----
  {
#pragma unroll
    for (int a = 0; a < 4; ++a)
#pragma unroll
      for (int q = 0; q < 4; ++q)
        sT[(ti + a) * LP + tj + q] = acc[a][q];             // materialize M
    __syncthreads();
    v8f a0 = wmma_mm_tile(sS, sT, t0 >> 2, t0 & 3, lane);
    v8f a1 = wmma_mm_tile(sS, sT, t1 >> 2, t1 & 3, lane);
    store_tile(sY, t0 >> 2, t0 & 3, lane, a0);
    store_tile(sY, t1 >> 2, t1 & 3, lane, a1);
    float ss = 0.f;
#pragma unroll
    for (int r = 0; r < 8; ++r) ss += a0[r] * a0[r] + a1[r] * a1[r];
    sRed[tid] = ss;
    __syncthreads();
    if (tid == 0) {
      float s = 0.f;
      for (int i = 0; i < 256; ++i) s += sRed[i];
      float nA = sqrtf(s);
      sScal[0] = nA;                               // keep ||A||_F for the end
      sScal[1] = (nA > 0.f) ? (1.0f / nA) : 0.0f;
    }
    __syncthreads();
    float rnA = sScal[1];
#pragma unroll
    for (int e = 0; e < 16; ++e) {
      int flat = e * 256 + tid;
      int r = flat >> 6, cc = flat & 63;
      sY[r * LP + cc] *= rnA;
      sZ[r * LP + cc] = (r == cc) ? 1.0f : 0.0f;
    }
    __syncthreads();
  }

  // ------------- coupled NS sqrt: T=1.5I-0.5 Z@Y; Y<-Y@T; Z<-T@Z -------------
  for (int it = 0; it < SQRT_ITERS; ++it) {
    v8f w0 = wmma_mm_tile(sZ, sY, t0 >> 2, t0 & 3, lane);
    v8f w1 = wmma_mm_tile(sZ, sY, t1 >> 2, t1 & 3, lane);
    store_T_tile(sT, t0 >> 2, t0 & 3, lane, w0);
    store_T_tile(sT, t1 >> 2, t1 & 3, lane, w1);
    __syncthreads();
    v8f r0, r1, r2, r3;                     // 4 jobs/wave: 16 Y-tiles + 16 Z-tiles
    {
      int j;
      j = wave * 4 + 0;
      r0 = (j < 16) ? wmma_mm_tile(sY, sT, j >> 2, j & 3, lane)
                    : wmma_mm_tile(sT, sZ, (j - 16) >> 2, (j - 16) & 3, lane);
      j = wave * 4 + 1;
      r1 = (j < 16) ? wmma_mm_tile(sY, sT, j >> 2, j & 3, lane)
                    : wmma_mm_tile(sT, sZ, (j - 16) >> 2, (j - 16) & 3, lane);
      j = wave * 4 + 2;
      r2 = (j < 16) ? wmma_mm_tile(sY, sT, j >> 2, j & 3, lane)
                    : wmma_mm_tile(sT, sZ, (j - 16) >> 2, (j - 16) & 3, lane);
      j = wave * 4 + 3;
      r3 = (j < 16) ? wmma_mm_tile(sY, sT, j >> 2, j & 3, lane)
                    : wmma_mm_tile(sT, sZ, (j - 16) >> 2, (j - 16) & 3, lane);
    }
    __syncthreads();
    {
      int j;
      j = wave * 4 + 0;
      if (j < 16) store_tile(sY, j >> 2, j & 3, lane, r0);
      else        store_tile(sZ, (j - 16) >> 2, (j - 16) & 3, lane, r0);
      j = wave * 4 + 1;
      if (j < 16) store_tile(sY, j >> 2, j & 3, lane, r1);
      else        store_tile(sZ, (j - 16) >> 2, (j - 16) & 3, lane, r1);
      j = wave * 4 + 2;
      if (j < 16) store_tile(sY, j >> 2, j & 3, lane, r2);
      else        store_tile(sZ, (j - 16) >> 2, (j - 16) & 3, lane, r2);
      j = wave * 4 + 3;
      if (j < 16) store_tile(sY, j >> 2, j & 3, lane, r3);
      else        store_tile(sZ, (j - 16) >> 2, (j - 16) & 3, lane, r3);
    }
    __syncthreads();
  }

  // ------------- sq = sqrt(normA) * (S @ Y); out = sq / max(||sq||, eps) -----
  {
    v8f f0 = wmma_mm_tile(sS, sY, t0 >> 2, t0 & 3, lane);
    v8f f1 = wmma_mm_tile(sS, sY, t1 >> 2, t1 & 3, lane);
    float ss = 0.f;
#pragma unroll
    for (int r = 0; r < 8; ++r) ss += f0[r] * f0[r] + f1[r] * f1[r];
    sRed[tid] = ss;
    __syncthreads();
    if (tid == 0) {
      float s = 0.f;
      for (int i = 0; i < 256; ++i) s += sRed[i];
      float fro = sqrtf(s);
      float snA = sqrtf(sScal[0]);
      float den = fmaxf(snA * fro, 1e-12f);
      sScal[1] = snA / den;
    }
    __syncthreads();
    float so = sScal[1];
    float* ob = out + (size_t)bat * 4096;
    const int lmod = lane & 15;
    const int lh   = lane >> 4;
    {
      int row = (t0 >> 2) * 16 + lh * 8, col = (t0 & 3) * 16 + lmod;
#pragma unroll
      for (int r = 0; r < 8; ++r) ob[(row + r) * 64 + col] = f0[r] * so;
    }
    {
      int row = (t1 >> 2) * 16 + lh * 8, col = (t1 & 3) * 16 + lmod;
#pragma unroll
      for (int r = 0; r < 8; ++r) ob[(row + r) * 64 + col] = f1[r] * so;
    }
  }
}

extern "C" void kernel_launch(void* const* d_in, const int* in_sizes, int n_in,
                              void* d_out, int out_size, void* d_ws, size_t ws_size,
                              hipStream_t stream) {
  (void)n_in; (void)d_ws; (void)ws_size; (void)out_size;
  const float* x = (const float*)d_in[0];
  float* out = (float*)d_out;
  const int nrows = 2048;                      // N (features) per batch
  int B = in_sizes[0] / (nrows * 64);          // 16
  sop_signsqrt_kernel<<<B, 256, 0, stream>>>(x, out, nrows);
}